// LSTM_6605659701960
// MI455X (gfx1250) — compile-verified
//
#include <hip/hip_runtime.h>

typedef float v2f __attribute__((ext_vector_type(2)));
typedef float v8f __attribute__((ext_vector_type(8)));
typedef unsigned int u32x4 __attribute__((ext_vector_type(4)));
typedef int i32x4 __attribute__((ext_vector_type(4)));
typedef int i32x8 __attribute__((ext_vector_type(8)));

#define S_LEN   512
#define BATCH   64
#define IDIM    1024
#define HDIM    1024
#define NBLOCKS 128
#define NTHR    256

// LDS x-tile: 32 rows, padded pitch. TDM pads 2 DWORDs every 256 DWORDs,
// so a 1024-float row occupies 1024 + 4*2 = 1032 floats in LDS.
#define XPITCH      1032
#define XBUF_FLOATS (32 * XPITCH)
#define XBUF_BYTES  (XBUF_FLOATS * 4)

__device__ __forceinline__ v2f ld2(const float* p) {
  return *(const v2f*)p;
}

// Tensor Data Mover: async copy of a 32x1024 f32 tile (row stride 1024 elems)
// from global memory into LDS at lds_byte_addr, inserting a 2-DWORD pad every
// 256 DWORDs (LDS row pitch becomes 1032 floats). D# per CDNA5 ISA ch.8.
__device__ __forceinline__ void tdm_load_x_tile(const float* src,
                                                unsigned lds_byte_addr) {
  unsigned long long ga = (unsigned long long)(uintptr_t)src;
  u32x4 g0;
  g0.x = 1u;                                       // count=1 (valid user D#)
  g0.y = lds_byte_addr;                            // lds_addr
  g0.z = (unsigned)(ga & 0xFFFFFFFFu);             // global_addr[31:0]
  g0.w = (unsigned)((ga >> 32) & 0x01FFFFFFu)      // global_addr[56:32]
         | (2u << 30);                             // type=2 ("image")
  i32x8 g1;
  g1.s0 = (int)((2u << 16)        // data_size = 4 bytes
              | (1u << 20)        // pad_enable
              | (7u << 22)        // pad_interval: 256 DWORDs
              | (1u << 25));      // pad_amount: 2 DWORDs
  g1.s1 = (int)(1024u << 16);     // tensor_dim0 = 1024
  g1.s2 = (int)(32u << 16);       // tensor_dim1 = 32
  g1.s3 = (int)(1024u << 16);     // tile_dim0 = 1024
  g1.s4 = 32;                     // tile_dim1 = 32 (tile_dim2 = 0)
  g1.s5 = 1024;                   // tensor_dim0_stride = 1024 elems
  g1.s6 = 0;
  g1.s7 = 0;
  i32x4 gz4 = {0, 0, 0, 0};       // groups 2/3 unused (<=2D tensor)
  i32x8 gz8 = {0, 0, 0, 0, 0, 0, 0, 0};
  __builtin_amdgcn_tensor_load_to_lds(g0, g1, gz4, gz4, gz8, 0);
}

__global__ void __launch_bounds__(NTHR, 1)
lstm_init_kernel(const float* __restrict__ h0, const float* __restrict__ c0,
                 float* __restrict__ h_state, float* __restrict__ c_state,
                 unsigned* __restrict__ sync) {
  int i = blockIdx.x * blockDim.x + threadIdx.x;
  if (i < BATCH * HDIM) {
    h_state[i] = h0[i];
    c_state[i] = c0[i];
  }
  if (i < 4) sync[i] = 0u;
}

__global__ void __launch_bounds__(NTHR, 1)
lstm_persistent_kernel(const float* __restrict__ x,
    const float* __restrict__ Wxi, const float* __restrict__ Wxf,
    const float* __restrict__ Wxo, const float* __restrict__ Wxg,
    const float* __restrict__ Whi, const float* __restrict__ Whf,
    const float* __restrict__ Who, const float* __restrict__ Whg,
    const float* __restrict__ bxi, const float* __restrict__ bxf,
    const float* __restrict__ bxo, const float* __restrict__ bxg,
    const float* __restrict__ bhi, const float* __restrict__ bhf,
    const float* __restrict__ bho, const float* __restrict__ bhg,
    float* __restrict__ out, float* h_state, float* c_state, unsigned* sync)
{
  // Double-buffered x tile staged by the TDM: lx[2][32][XPITCH]
  extern __shared__ float lx[];
  // Gate tiles: [gate][rowtile-in-half][row][col], padded to 17 floats/row
  __shared__ float lg[4][2][16][17];

  const int tid  = threadIdx.x;
  const int wave = tid >> 5;      // 0..7 (wave32)
  const int lane = tid & 31;
  const int lr   = lane & 15;     // row/col within half-wave
  const int lh   = lane >> 4;     // half-wave index

  const int wg    = blockIdx.x;   // 0..127
  const int jt    = wg >> 1;      // h-column tile 0..63 (16 cols each)
  const int rhalf = wg & 1;       // batch rows [32*rhalf, 32*rhalf+32)
  const int gate  = wave >> 1;    // 0:i 1:f 2:o 3:g
  const int rt    = (rhalf << 1) | (wave & 1);  // rowtile 0..3
  const int m0    = rt << 4;

  const float* Wxp = (gate == 0) ? Wxi : (gate == 1) ? Wxf : (gate == 2) ? Wxo : Wxg;
  const float* Whp = (gate == 0) ? Whi : (gate == 1) ? Whf : (gate == 2) ? Who : Whg;

  // B-matrix fragments: B = W^T, lane(v in {0,1}) holds B[v+2*lh][lr]
  // -> contiguous float2 at W[(jt*16+lr)*ld + k0 + 2*lh]
  const int ncol = (jt << 4) + lr;
  const float* wxrow = Wxp + (size_t)ncol * IDIM + (lh << 1);
  const float* whrow = Whp + (size_t)ncol * HDIM + (lh << 1);

  // A-matrix fragments: lane holds A[lr][k0 + 2*lh + {0,1}]
  const int brow = m0 + lr;                        // global batch row
  const int arow = ((wave & 1) << 4) + lr;         // row within staged tile
  const float* hrow = h_state + (size_t)brow * HDIM + (lh << 1);

  const unsigned lx_lds = (unsigned)(uintptr_t)(void*)lx;
  const float* xtile_src = x + (size_t)(rhalf << 5) * IDIM;  // + s*B*I per step

  // Prologue: stage x tile for step 0 into buffer 0
  if (wave == 0)
    tdm_load_x_tile(xtile_src, lx_lds);

  for (int s = 0; s < S_LEN; ++s) {
    // Make the staged x tile for this step visible
    if (wave == 0) __builtin_amdgcn_s_wait_tensorcnt((short)0);
    __syncthreads();

    // Kick off the DMA for the next step's x tile (overlaps this step's GEMM)
    if (wave == 0 && (s + 1) < S_LEN)
      tdm_load_x_tile(xtile_src + (size_t)(s + 1) * (BATCH * IDIM),
                      lx_lds + (unsigned)(((s + 1) & 1) * XBUF_BYTES));

    // ---- x-side GEMM: A from LDS (padded pitch), B (Wx^T) from global ----
    v8f accx0 = {0.f,0.f,0.f,0.f,0.f,0.f,0.f,0.f};
    v8f accx1 = {0.f,0.f,0.f,0.f,0.f,0.f,0.f,0.f};
    const float* axbase = lx + (size_t)(s & 1) * XBUF_FLOATS
                             + (size_t)arow * XPITCH + (lh << 1);
#pragma unroll
    for (int kb = 0; kb < 4; ++kb) {           // 4 segments of 256 (pad-aware)
      const float* ax = axbase + kb * 258;     // 256 data + 2 pad floats
      const float* bx = wxrow + (kb << 8);
#pragma unroll 4
      for (int k = 0; k < 256; k += 8) {
        v2f a0 = ld2(ax + k);
        v2f b0 = ld2(bx + k);
        accx0 = __builtin_amdgcn_wmma_f32_16x16x4_f32(
            false, a0, false, b0, (short)0, accx0, false, false);
        v2f a1 = ld2(ax + k + 4);
        v2f b1 = ld2(bx + k + 4);
        accx1 = __builtin_amdgcn_wmma_f32_16x16x4_f32(
            false, a1, false, b1, (short)0, accx1, false, false);
      }
    }

    // ---- h-side GEMM: A (h_state) and B (Wh^T) from global/L2 ----
    v8f acch0 = {0.f,0.f,0.f,0.f,0.f,0.f,0.f,0.f};
    v8f acch1 = {0.f,0.f,0.f,0.f,0.f,0.f,0.f,0.f};
#pragma unroll 4
    for (int k = 0; k < HDIM; k += 8) {
      v2f a0 = ld2(hrow + k);
      v2f b0 = ld2(whrow + k);
      acch0 = __builtin_amdgcn_wmma_f32_16x16x4_f32(
          false, a0, false, b0, (short)0, acch0, false, false);
      v2f a1 = ld2(hrow + k + 4);
      v2f b1 = ld2(whrow + k + 4);
      acch1 = __builtin_amdgcn_wmma_f32_16x16x4_f32(
          false, a1, false, b1, (short)0, acch1, false, false);
    }

    v8f acc = (accx0 + accx1) + (acch0 + acch1);

    // C/D layout: VGPR p, half-wave lh -> row = p + 8*lh, col = lr
#pragma unroll
    for (int p = 0; p < 8; ++p)
      lg[gate][wave & 1][p + (lh << 3)][lr] = acc[p];

    __syncthreads();

    // Elementwise LSTM cell update for this block's 32 batch rows x 16 cols
    for (int e = tid; e < 512; e += NTHR) {
      int row = e >> 4;              // 0..31
      int col = e & 15;
      int b   = (rhalf << 5) + row;  // batch index
      int j   = (jt << 4) + col;     // hidden index
      float gi = lg[0][row >> 4][row & 15][col] + bxi[j] + bhi[j];
      float gf = lg[1][row >> 4][row & 15][col] + bxf[j] + bhf[j];
      float go = lg[2][row >> 4][row & 15][col] + bxo[j] + bho[j];
      float gg = lg[3][row >> 4][row & 15][col] + bxg[j] + bhg[j];
      float it = 1.f / (1.f + __expf(-gi));
      float ft = 1.f / (1.f + __expf(-gf));
      float ot = 1.f / (1.f + __expf(-go));
      float gt = tanhf(gg);
      size_t off = (size_t)b * HDIM + j;
      float cn = ft * c_state[off] + it * gt;
      float hn = ot * tanhf(cn);
      c_state[off] = cn;
      h_state[off] = hn;
      out[(size_t)s * (BATCH * HDIM) + off] = hn;
      if (s == S_LEN - 1) {
        out[(size_t)S_LEN * (BATCH * HDIM) + off] = hn;
        out[(size_t)S_LEN * (BATCH * HDIM) + (BATCH * HDIM) + off] = cn;
      }
    }

    // Grid-wide barrier: release h_state/c_state, sense via generation counter
    __threadfence();
    __syncthreads();
    if (tid == 0) {
      unsigned target  = (unsigned)(s + 1);
      unsigned arrived = atomicAdd(&sync[0], 1u) + 1u;
      if (arrived == (unsigned)NBLOCKS) {
        sync[0] = 0u;
        __threadfence();
        atomicExch(&sync[1], target);
      } else {
        volatile unsigned* gen = &sync[1];
        while (*gen < target) __builtin_amdgcn_s_sleep(2);
      }
      __threadfence();
    }
    __syncthreads();
  }
}

extern "C" void kernel_launch(void* const* d_in, const int* in_sizes, int n_in,
                              void* d_out, int out_size, void* d_ws, size_t ws_size,
                              hipStream_t stream) {
  const float* x   = (const float*)d_in[0];
  const float* h0  = (const float*)d_in[1];
  const float* c0  = (const float*)d_in[2];
  const float* Wxi = (const float*)d_in[3];
  const float* Wxf = (const float*)d_in[4];
  const float* Wxo = (const float*)d_in[5];
  const float* Wxg = (const float*)d_in[6];
  const float* Whi = (const float*)d_in[7];
  const float* Whf = (const float*)d_in[8];
  const float* Who = (const float*)d_in[9];
  const float* Whg = (const float*)d_in[10];
  const float* bxi = (const float*)d_in[11];
  const float* bxf = (const float*)d_in[12];
  const float* bxo = (const float*)d_in[13];
  const float* bxg = (const float*)d_in[14];
  const float* bhi = (const float*)d_in[15];
  const float* bhf = (const float*)d_in[16];
  const float* bho = (const float*)d_in[17];
  const float* bhg = (const float*)d_in[18];

  float* out      = (float*)d_out;
  float* h_state  = (float*)d_ws;                        // 64*1024 f32
  float* c_state  = h_state + BATCH * HDIM;              // 64*1024 f32
  unsigned* sync  = (unsigned*)(c_state + BATCH * HDIM); // 4 u32

  lstm_init_kernel<<<(BATCH * HDIM + NTHR - 1) / NTHR, NTHR, 0, stream>>>(
      h0, c0, h_state, c_state, sync);

  lstm_persistent_kernel<<<NBLOCKS, NTHR, 2 * XBUF_BYTES, stream>>>(
      x, Wxi, Wxf, Wxo, Wxg, Whi, Whf, Who, Whg,
      bxi, bxf, bxo, bxg, bhi, bhf, bho, bhg,
      out, h_state, c_state, sync);
}